// AttentionConv_86904368267928
// MI455X (gfx1250) — compile-verified
//
#include <hip/hip_runtime.h>
#include <hip/hip_bf16.h>

// ---------------------------------------------------------------------------
// Problem constants (from reference)
// ---------------------------------------------------------------------------
#define BB   2
#define CIN  256
#define HH   48
#define WW   48
#define OCC  256
#define GG   4
#define GOC  64          // OC / G
#define C2   32          // GOC / 2
#define KHW  7
#define PAD  3
#define HP   54          // H + 2*PAD
#define WP   54
#define TC   768         // 3 * OC  (q | k | v channel blocks)
#define PP   2304        // Hn * Wn = 48*48
// Intermediate y layout: POSITION-MAJOR  yt[b][hp][wp][tc]
#define YT_FLOATS ((size_t)BB * HP * WP * TC)

// LDS geometry for the GEMM
#define A_STRIDE 36                    // 64 x 36, rows 16B aligned, pairs 8B aligned
#define B_STRIDE 132                   // 32 x 132, rows 16B aligned
#define BUF_FLOATS (64 * A_STRIDE + 32 * B_STRIDE)   // 6528 per buffer

typedef float v2f __attribute__((ext_vector_type(2)));
typedef float v8f __attribute__((ext_vector_type(8)));

// ---------------------------------------------------------------------------
// gfx1250 async global->LDS copy (ASYNCcnt-tracked, bypasses VGPR staging).
// VDST operand is the per-lane LDS byte address; a generic pointer to a
// __shared__ object carries the LDS offset in its low 32 bits (flat shared
// aperture encoding), so truncation recovers the DS address.
// ---------------------------------------------------------------------------
__device__ __forceinline__ void async_load_b128(void* lds_ptr, const void* gptr)
{
    asm volatile("global_load_async_to_lds_b128 %0, %1, off"
                 :: "v"((uint32_t)(uintptr_t)lds_ptr), "v"(gptr)
                 : "memory");
}
// Async completions retire in issue order, so waiting for ASYNCcnt <= N
// releases everything issued before the most recent N operations.
template <int N>
__device__ __forceinline__ void wait_async_le()
{
    asm volatile("s_wait_asynccnt %0" :: "n"(N) : "memory");
}

// ---------------------------------------------------------------------------
// Kernel 1: 1x1 conv as fp32 WMMA GEMM, double-buffered async staging.
//   Per batch:  C[768 x 2304] = W[768 x 256] @ X[256 x 2304]
//   Block = 256 threads = 8 wave32; tile 64(M) x 128(N); wave = 16 x 64.
//   Pipeline: stage(next buf) -> s_wait_asynccnt 6 -> barrier -> 32x WMMA,
//   so the next chunk's 6 wave-level B128 transfers fly behind the math.
//   Output written position-major into padded yt (borders pre-zeroed).
// ---------------------------------------------------------------------------
__global__ __launch_bounds__(256)
void conv_gemm_wmma(const float* __restrict__ x,
                    const float* __restrict__ w,
                    float* __restrict__ yt)
{
    // smem: 2 x (As[64][36] + Bs[32][132]) = 13056 f  |  Cs[128][65] = 8320 f
    __shared__ float smem[2 * BUF_FLOATS];

    const int t    = threadIdx.x;
    const int lane = t & 31;
    const int wv   = t >> 5;          // 0..7
    const int wm   = wv & 3;          // M-wave 0..3
    const int wn   = wv >> 2;         // N-wave 0..1
    const int lh   = lane >> 4;       // lane half (K split)
    const int ll   = lane & 15;       // lane low (M / N index)

    const int b  = blockIdx.z;
    const int m0 = blockIdx.y * 64;   // out-channel tile
    const int n0 = blockIdx.x * 128;  // spatial tile

    const float* xb = x + (size_t)b * CIN * PP;

    // Issue one chunk's staging: 2 (A) + 4 (B) async b128 ops per wave.
    auto stage = [&](int buf, int kc) {
        float* A = smem + buf * BUF_FLOATS;
        float* B = A + 64 * A_STRIDE;
        #pragma unroll
        for (int i = 0; i < 2; ++i) {            // A tile 64x32, contiguous in K
            const int idx = (i * 256 + t) * 4;
            const int mo = idx >> 5, kk = idx & 31;
            async_load_b128(&A[mo * A_STRIDE + kk],
                            &w[(m0 + mo) * CIN + kc + kk]);
        }
        #pragma unroll
        for (int i = 0; i < 4; ++i) {            // B tile 32x128, contiguous in N
            const int idx = (i * 256 + t) * 4;
            const int kk = idx >> 7, nn = idx & 127;
            async_load_b128(&B[kk * B_STRIDE + nn],
                            &xb[(size_t)(kc + kk) * PP + n0 + nn]);
        }
    };

    v8f acc[4];
    #pragma unroll
    for (int j = 0; j < 4; ++j)
        #pragma unroll
        for (int i = 0; i < 8; ++i) acc[j][i] = 0.0f;

    stage(0, 0);                                 // prologue

    #pragma unroll
    for (int c = 0; c < 8; ++c) {                // 8 K-chunks of 32
        const int cur = c & 1;
        if (c < 7) {
            stage(1 - cur, (c + 1) * 32);        // overlap next chunk
            wait_async_le<6>();                  // current chunk landed
        } else {
            wait_async_le<0>();
        }
        __syncthreads();

        const float* As = smem + cur * BUF_FLOATS;
        const float* Bs = As + 64 * A_STRIDE;

        #pragma unroll
        for (int ks = 0; ks < 8; ++ks) {         // 8 wmma K-steps of 4
            const int kb = ks * 4 + lh * 2;      // ISA 16x4 f32 A layout:
            v2f a;                               // lane<16: K0,K1 ; lane>=16: K2,K3
            a.x = As[(wm * 16 + ll) * A_STRIDE + kb];
            a.y = As[(wm * 16 + ll) * A_STRIDE + kb + 1];
            #pragma unroll
            for (int j = 0; j < 4; ++j) {
                const int n = wn * 64 + j * 16 + ll;
                v2f bv;                          // mirrored 4x16 B layout
                bv.x = Bs[kb * B_STRIDE + n];
                bv.y = Bs[(kb + 1) * B_STRIDE + n];
                acc[j] = __builtin_amdgcn_wmma_f32_16x16x4_f32(
                    /*neg_a=*/false, a, /*neg_b=*/false, bv,
                    /*c_mod=*/(short)0, acc[j],
                    /*reuse_a=*/false, /*reuse_b=*/false);
            }
        }
        __syncthreads();                          // done reading 'cur' before
    }                                             // it is re-staged (c+2)

    // Transpose through LDS so global stores are channel-contiguous
    float* Cs = smem;  // [128][65], n-major
    #pragma unroll
    for (int j = 0; j < 4; ++j) {
        const int nloc = wn * 64 + j * 16 + ll;
        #pragma unroll
        for (int i = 0; i < 8; ++i) {
            const int oloc = wm * 16 + lh * 8 + i;   // ISA 16x16 f32 C layout
            Cs[nloc * 65 + oloc] = acc[j][i];
        }
    }
    __syncthreads();

    for (int idx = t; idx < 128 * 64; idx += 256) {
        const int nloc = idx >> 6, oloc = idx & 63;
        const int s  = n0 + nloc;            // flat spatial 0..2303
        const int rr = s / 48, cc = s % 48;  // -> padded (rr+3, cc+3)
        yt[((size_t)(b * HP + rr + PAD) * WP + (cc + PAD)) * TC + m0 + oloc] =
            Cs[nloc * 65 + oloc];
    }
}

// ---------------------------------------------------------------------------
// Kernel 2: permuted local attention (bug-faithful to reference reshape).
// One workgroup per (b, g, output row r).  For fixed r:
//   gsel = 2g + (r>=24)       (uniform), selects channel half + rpe table
//   source pos for s0 = 2p mod P, s1 = s0 + 1
// logit(p,k) = q_lo . Kraw[sa,j0] + q_hi . Kraw[sb,j1] + qlo.R(j0) + qhi.R(j1)
// ---------------------------------------------------------------------------
__global__ __launch_bounds__(256)
void attn_kernel(const float* __restrict__ yt,
                 const float* __restrict__ rpe_h,
                 const float* __restrict__ rpe_w,
                 float* __restrict__ out)
{
    const int r = blockIdx.x;   // 0..47
    const int g = blockIdx.y;   // 0..3
    const int b = blockIdx.z;
    const int t = threadIdx.x;

    __shared__ float Rtab[7 * 32];       // rpe slice for this workgroup
    __shared__ float QR[48 * 8 * 2];     // q . rpe  per (cp, tt, half)
    __shared__ float Slog[48 * 50];      // logits -> softmax weights
    __shared__ float Sout[64 * 49];      // transpose buffer for output

    const int  dlt   = (r >= 24) ? 1 : 0;
    const int  gsel  = 2 * g + dlt;          // 0..7
    const bool hsel  = (gsel < GG);          // low-half + rpe_h, else high + rpe_w
    const int  chabs = hsel ? (OCC + gsel * GOC)
                            : (OCC + (gsel - GG) * GOC + C2);
    const float* ybase = yt + (size_t)b * HP * WP * TC;

    // prefetch q row (-> global_prefetch_b8)
    if (t < 48)
        __builtin_prefetch(ybase + ((size_t)(r + PAD) * WP + (t + PAD)) * TC + g * GOC, 0, 1);

    if (t < 7 * 32) {
        const int tt = t >> 5, cc = t & 31;
        Rtab[t] = hsel ? rpe_h[gsel * 224 + tt * 32 + cc]
                       : rpe_w[(gsel - GG) * 224 + tt * 32 + cc];
    }
    __syncthreads();

    // q . rpe pre-dots: 48 positions x 7 window indices x 2 halves
    for (int id = t; id < 48 * 14; id += 256) {
        const int cp   = id / 14;
        const int rem  = id % 14;
        const int tt   = rem >> 1;
        const int half = rem & 1;
        const float* q  = ybase + ((size_t)(r + PAD) * WP + (cp + PAD)) * TC
                          + g * GOC + half * C2;
        const float* rt = &Rtab[tt * 32];
        float s = 0.f;
        #pragma unroll
        for (int cc = 0; cc < 32; ++cc) s += q[cc] * rt[cc];
        QR[(cp * 8 + tt) * 2 + half] = s;
    }
    __syncthreads();

    // ---- logits (permuted key gather) ----
    const int p0 = 2 * (r * 48) - dlt * PP;   // source flat pos for cp=0
    for (int id = t; id < 48 * 49; id += 256) {
        const int cp = id / 49;
        const int k  = id % 49;
        const int pp0 = p0 + 2 * cp;              // s0 source position
        const int r0 = pp0 / 48, c0 = pp0 % 48;
        const int r1 = r0 + (c0 == 47);           // s1 = s0 + 1
        const int c1 = (c0 == 47) ? 0 : (c0 + 1);

        int j0, j1, sa, sb;
        if (k <= 23)      { sa = 0; j0 = 2 * k;      sb = 0; j1 = 2 * k + 1;  }
        else if (k == 24) { sa = 0; j0 = 48;         sb = 1; j1 = 0;          }
        else              { sa = 1; j0 = 2 * k - 49; sb = 1; j1 = 2 * k - 48; }

        const int rA = sa ? r1 : r0, cA = sa ? c1 : c0;
        const int rB = sb ? r1 : r0, cB = sb ? c1 : c0;
        const int khA = j0 / 7, kwA = j0 % 7;
        const int khB = j1 / 7, kwB = j1 % 7;

        const float* q  = ybase + ((size_t)(r + PAD) * WP + (cp + PAD)) * TC + g * GOC;
        const float* kA = ybase + ((size_t)(rA + khA) * WP + (cA + kwA)) * TC + chabs;
        const float* kB = ybase + ((size_t)(rB + khB) * WP + (cB + kwB)) * TC + chabs;

        float lo = 0.f, hi = 0.f;
        #pragma unroll
        for (int cc = 0; cc < 32; ++cc) {
            lo += q[cc]      * kA[cc];
            hi += q[32 + cc] * kB[cc];
        }
        const int tA = hsel ? khA : kwA;
        const int tB = hsel ? khB : kwB;
        Slog[cp * 50 + k] = lo + hi + QR[(cp * 8 + tA) * 2 + 0]
                                    + QR[(cp * 8 + tB) * 2 + 1];
    }
    __syncthreads();

    // ---- softmax over 49 (one thread per position) ----
    if (t < 48) {
        float m = -1e30f;
        #pragma unroll
        for (int k = 0; k < 49; ++k) m = fmaxf(m, Slog[t * 50 + k]);
        float s = 0.f;
        #pragma unroll
        for (int k = 0; k < 49; ++k) {
            const float e = __expf(Slog[t * 50 + k] - m);
            Slog[t * 50 + k] = e;
            s += e;
        }
        const float inv = 1.0f / s;
        #pragma unroll
        for (int k = 0; k < 49; ++k) Slog[t * 50 + k] *= inv;
    }
    __syncthreads();

    // ---- att @ V (value is NOT permuted: local window at p) ----
    for (int id = t; id < 48 * 64; id += 256) {
        const int cp = id >> 6;
        const int c  = id & 63;
        const float* vb   = ybase + 2 * OCC + g * GOC + c;   // v block channel
        const float* wrow = &Slog[cp * 50];
        float acc = 0.f;
        #pragma unroll
        for (int kh = 0; kh < 7; ++kh)
            #pragma unroll
            for (int kw = 0; kw < 7; ++kw)
                acc += wrow[kh * 7 + kw] *
                       vb[((size_t)(r + kh) * WP + (cp + kw)) * TC];
        Sout[c * 49 + cp] = acc;
    }
    __syncthreads();

    // coalesced output store: out[b][g*64+c][r][cp]
    const size_t ob = ((size_t)b * OCC + g * GOC) * PP + (size_t)r * 48;
    for (int id = t; id < 48 * 64; id += 256) {
        const int c = id / 48, pcol = id % 48;
        out[ob + (size_t)c * PP + pcol] = Sout[c * 49 + pcol];
    }
}

// ---------------------------------------------------------------------------
// Launcher
// ---------------------------------------------------------------------------
extern "C" void kernel_launch(void* const* d_in, const int* in_sizes, int n_in,
                              void* d_out, int out_size, void* d_ws, size_t ws_size,
                              hipStream_t stream)
{
    const float* x  = (const float*)d_in[0];   // (2,256,48,48)
    const float* w  = (const float*)d_in[1];   // (768,256)
    const float* rh = (const float*)d_in[2];   // (4,1,7,1,32)
    const float* rw = (const float*)d_in[3];   // (4,1,1,7,32)
    float* out = (float*)d_out;                // (2,256,48,48)
    float* yt  = (float*)d_ws;                 // (2,54,54,768) position-major

    // zero padded borders (graph-capture legal)
    hipMemsetAsync(d_ws, 0, YT_FLOATS * sizeof(float), stream);

    dim3 gg(PP / 128, TC / 64, BB);            // 18 x 12 x 2
    conv_gemm_wmma<<<gg, 256, 0, stream>>>(x, w, yt);

    dim3 ga(48, GG, BB);                       // 48 x 4 x 2
    attn_kernel<<<ga, 256, 0, stream>>>(yt, rh, rw, out);
}